// ToPo_Agg_558345749113
// MI455X (gfx1250) — compile-verified
//
#include <hip/hip_runtime.h>
#include <math.h>

#define N_NODES 100000
#define N_EDGES 1600000
#define HID     64
#define T0_DIM  128
#define T1_DIM  256
#define SEM_HID 128

typedef float v2f __attribute__((ext_vector_type(2)));
typedef float v8f __attribute__((ext_vector_type(8)));

// ------------------------------------------------------------------ fills ---
__global__ void fill_kernel(float* __restrict__ p, long long n, float v) {
    long long i = (long long)blockIdx.x * blockDim.x + threadIdx.x;
    long long st = (long long)gridDim.x * blockDim.x;
    for (; i < n; i += st) p[i] = v;
}

// --------------------------------------------- fp32 WMMA GEMM  C = A * B ---
// C[N x 64] = A[N x K] * B[K x 64].  One wave per 16x16 C tile, 4 waves/block
// cover the 64 output columns.  act=1 -> C = tanh(C + bias[col]).
// V_WMMA_F32_16X16X4_F32 operand layout (ISA 7.12.2):
//   A 16x4 : lane holds row (lane&15), k-pair base (lane>>4)*2 in 2 VGPRs
//   B 4x16 : lane holds col (lane&15), same k-pair base in 2 VGPRs
//   C 16x16: 8 VGPRs, row = (lane>>4)*8 + m, col = lane&15
__global__ void gemm_f32_wmma(const float* __restrict__ A,
                              const float* __restrict__ B,
                              float* __restrict__ C, int K,
                              const float* __restrict__ bias, int act) {
    const int lane = threadIdx.x & 31;
    const int wave = threadIdx.x >> 5;
    const int row0 = blockIdx.x * 16;
    const int col0 = wave * 16;
    const int r    = lane & 15;
    const int kb   = (lane >> 4) * 2;

    v8f acc = {};
    const float* __restrict__ Arow = A + (size_t)(row0 + r) * K;
    for (int k = 0; k < K; k += 4) {
        v2f a, b;
        a.x = Arow[k + kb];
        a.y = Arow[k + kb + 1];
        b.x = B[(size_t)(k + kb)     * HID + col0 + r];
        b.y = B[(size_t)(k + kb + 1) * HID + col0 + r];
        acc = __builtin_amdgcn_wmma_f32_16x16x4_f32(
            false, a, false, b, (short)0, acc, false, false);
    }

    const int crow = row0 + (lane >> 4) * 8;
    const int ccol = col0 + r;
#pragma unroll
    for (int m = 0; m < 8; ++m) {
        float v = acc[m];
        if (act) v = tanhf(v + bias[ccol]);
        C[(size_t)(crow + m) * HID + ccol] = v;
    }
}

// lin1: A = concat(f0, f1) along feature dim (K = 128), epilogue tanh(.+b).
__global__ void gemm_concat_wmma(const float* __restrict__ A0,
                                 const float* __restrict__ A1,
                                 const float* __restrict__ B,
                                 float* __restrict__ C,
                                 const float* __restrict__ bias) {
    const int lane = threadIdx.x & 31;
    const int wave = threadIdx.x >> 5;
    const int row0 = blockIdx.x * 16;
    const int col0 = wave * 16;
    const int r    = lane & 15;
    const int kb   = (lane >> 4) * 2;

    v8f acc = {};
    const float* __restrict__ R0 = A0 + (size_t)(row0 + r) * HID;
    const float* __restrict__ R1 = A1 + (size_t)(row0 + r) * HID;
    for (int k = 0; k < 2 * HID; k += 4) {
        int kk = k + kb;                        // kk, kk+1 same side of 64
        v2f a, b;
        a.x = (kk < HID) ? R0[kk]     : R1[kk - HID];
        a.y = (kk < HID) ? R0[kk + 1] : R1[kk + 1 - HID];
        b.x = B[(size_t)kk       * HID + col0 + r];
        b.y = B[(size_t)(kk + 1) * HID + col0 + r];
        acc = __builtin_amdgcn_wmma_f32_16x16x4_f32(
            false, a, false, b, (short)0, acc, false, false);
    }

    const int crow = row0 + (lane >> 4) * 8;
    const int ccol = col0 + r;
#pragma unroll
    for (int m = 0; m < 8; ++m)
        C[(size_t)(crow + m) * HID + ccol] = tanhf(acc[m] + bias[ccol]);
}

// ------------------------------------------------------------- GCN scatter --
// out[dst[e], c] += support[src[e], c] * val[e]; 64 consecutive lanes = one
// edge's 64 channels -> coalesced gather + coalesced f32 atomics.
__global__ void gcn_scatter(const float* __restrict__ sup,
                            const int* __restrict__ src,
                            const int* __restrict__ dst,
                            const float* __restrict__ val,
                            float* __restrict__ out) {
    long long total = (long long)N_EDGES * HID;
    long long i  = (long long)blockIdx.x * blockDim.x + threadIdx.x;
    long long st = (long long)gridDim.x * blockDim.x;
    for (; i < total; i += st) {
        int e = (int)(i >> 6);
        int c = (int)(i & 63);
        atomicAdd(&out[(size_t)dst[e] * HID + c],
                  sup[(size_t)src[e] * HID + c] * val[e]);
    }
}

// --------------------------------------------------------------- epilogues --
__global__ void bias_tanh_inplace(float* __restrict__ x,
                                  const float* __restrict__ bias) {
    long long total = (long long)N_NODES * HID;
    long long i  = (long long)blockIdx.x * blockDim.x + threadIdx.x;
    long long st = (long long)gridDim.x * blockDim.x;
    for (; i < total; i += st)
        x[i] = tanhf(x[i] + bias[i & 63]);
}

// ----------------------------------------------------------------- GAT ------
__global__ void gat_eler(const float* __restrict__ h,
                         const float* __restrict__ al,
                         const float* __restrict__ ar,
                         float* __restrict__ el, float* __restrict__ er) {
    int n = blockIdx.x * blockDim.x + threadIdx.x;
    if (n >= N_NODES) return;
    float sl = 0.f, sr = 0.f;
    const float* row = h + (size_t)n * HID;
#pragma unroll
    for (int d = 0; d < HID; ++d) { float v = row[d]; sl += v * al[d]; sr += v * ar[d]; }
    el[n] = sl; er[n] = sr;
}

__device__ inline void atomicMaxF(float* addr, float val) {
    unsigned int* ua = (unsigned int*)addr;
    unsigned int old = __float_as_uint(*addr);
    while (__uint_as_float(old) < val) {
        unsigned int assumed = old;
        old = atomicCAS(ua, assumed, __float_as_uint(val));
        if (old == assumed) break;
    }
}

__global__ void gat_edge1(const float* __restrict__ el, const float* __restrict__ er,
                          const int* __restrict__ src, const int* __restrict__ dst,
                          float* __restrict__ ebuf, float* __restrict__ emax) {
    int i  = blockIdx.x * blockDim.x + threadIdx.x;
    int st = gridDim.x * blockDim.x;
    for (; i < N_EDGES; i += st) {
        float x = el[src[i]] + er[dst[i]];
        x = (x > 0.f) ? x : 0.2f * x;          // leaky_relu 0.2
        ebuf[i] = x;
        atomicMaxF(&emax[dst[i]], x);
    }
}

__global__ void gat_edge2(float* __restrict__ ebuf, const int* __restrict__ dst,
                          const float* __restrict__ emax, float* __restrict__ denom) {
    int i  = blockIdx.x * blockDim.x + threadIdx.x;
    int st = gridDim.x * blockDim.x;
    for (; i < N_EDGES; i += st) {
        int d = dst[i];
        float ee = expf(ebuf[i] - emax[d]);
        ebuf[i] = ee;
        atomicAdd(&denom[d], ee);
    }
}

__global__ void gat_scatter(const float* __restrict__ h,
                            const int* __restrict__ src, const int* __restrict__ dst,
                            const float* __restrict__ ebuf,
                            const float* __restrict__ denom,
                            float* __restrict__ out) {
    long long total = (long long)N_EDGES * HID;
    long long i  = (long long)blockIdx.x * blockDim.x + threadIdx.x;
    long long st = (long long)gridDim.x * blockDim.x;
    for (; i < total; i += st) {
        int e = (int)(i >> 6);
        int c = (int)(i & 63);
        int d = dst[e];
        float alpha = ebuf[e] / fmaxf(denom[d], 1e-12f);
        atomicAdd(&out[(size_t)d * HID + c],
                  alpha * h[(size_t)src[e] * HID + c]);
    }
}

// -------------------------------------------------- semantic attention ------
// Only mean_n w[n,k] matters for beta -> reduce directly into wsum[2].
__global__ void sem_kernel(const float* __restrict__ l, const float* __restrict__ ff,
                           const float* __restrict__ W1, const float* __restrict__ b1,
                           const float* __restrict__ W2, float* __restrict__ wsum) {
    __shared__ float red0[256];
    __shared__ float red1[256];
    int n = blockIdx.x * blockDim.x + threadIdx.x;
    float s0 = 0.f, s1 = 0.f;
    if (n < N_NODES) {
        float z[HID];
#pragma unroll
        for (int d = 0; d < HID; ++d) z[d] = l[(size_t)n * HID + d];
        for (int j = 0; j < SEM_HID; ++j) {
            float s = b1[j];
#pragma unroll
            for (int d = 0; d < HID; ++d) s += z[d] * W1[d * SEM_HID + j];
            s0 += tanhf(s) * W2[j];
        }
#pragma unroll
        for (int d = 0; d < HID; ++d) z[d] = ff[(size_t)n * HID + d];
        for (int j = 0; j < SEM_HID; ++j) {
            float s = b1[j];
#pragma unroll
            for (int d = 0; d < HID; ++d) s += z[d] * W1[d * SEM_HID + j];
            s1 += tanhf(s) * W2[j];
        }
    }
    red0[threadIdx.x] = s0; red1[threadIdx.x] = s1;
    __syncthreads();
    for (int s = 128; s > 0; s >>= 1) {
        if ((int)threadIdx.x < s) {
            red0[threadIdx.x] += red0[threadIdx.x + s];
            red1[threadIdx.x] += red1[threadIdx.x + s];
        }
        __syncthreads();
    }
    if (threadIdx.x == 0) {
        atomicAdd(&wsum[0], red0[0]);
        atomicAdd(&wsum[1], red1[0]);
    }
}

__global__ void beta_kernel(const float* __restrict__ wsum, float* __restrict__ beta) {
    float b0 = wsum[0] / (float)N_NODES;
    float b1 = wsum[1] / (float)N_NODES;
    float m  = fmaxf(b0, b1);
    float e0 = expf(b0 - m), e1 = expf(b1 - m);
    float d  = e0 + e1;
    beta[0] = e0 / d;
    beta[1] = e1 / d;
}

__global__ void final_kernel(const float* __restrict__ l, const float* __restrict__ ff,
                             const float* __restrict__ beta, float* __restrict__ out) {
    long long total = (long long)N_NODES * HID;
    long long i  = (long long)blockIdx.x * blockDim.x + threadIdx.x;
    long long st = (long long)gridDim.x * blockDim.x;
    float b0 = beta[0], b1 = beta[1];
    for (; i < total; i += st) out[i] = b0 * l[i] + b1 * ff[i];
}

// ------------------------------------------------------------------ launch --
extern "C" void kernel_launch(void* const* d_in, const int* in_sizes, int n_in,
                              void* d_out, int out_size, void* d_ws, size_t ws_size,
                              hipStream_t stream) {
    const float* feature  = (const float*)d_in[0];
    const float* attr0    = (const float*)d_in[1];
    const float* attr1    = (const float*)d_in[2];
    const float* adj0_val = (const float*)d_in[3];
    const float* adj1_val = (const float*)d_in[4];
    const int*   adj0_src = (const int*)d_in[5];
    const int*   adj0_dst = (const int*)d_in[6];
    const int*   adj1_src = (const int*)d_in[7];
    const int*   adj1_dst = (const int*)d_in[8];
    const int*   sim_src  = (const int*)d_in[9];
    const int*   sim_dst  = (const int*)d_in[10];
    const float* gcn0_W   = (const float*)d_in[11];
    const float* gcn0_b   = (const float*)d_in[12];
    const float* gcn1_W   = (const float*)d_in[13];
    const float* gcn1_b   = (const float*)d_in[14];
    const float* gat_W    = (const float*)d_in[15];
    const float* gat_al   = (const float*)d_in[16];
    const float* gat_ar   = (const float*)d_in[17];
    const float* gat_bias = (const float*)d_in[18];
    const float* sem_W1   = (const float*)d_in[19];
    const float* sem_b1   = (const float*)d_in[20];
    const float* sem_W2   = (const float*)d_in[21];
    const float* lin1_W   = (const float*)d_in[22];
    const float* lin1_b   = (const float*)d_in[23];
    float* out = (float*)d_out;

    // workspace carve (floats); big buffers reused: supA->h, supB->l
    float* ws    = (float*)d_ws;
    size_t NH    = (size_t)N_NODES * HID;
    float* supA  = ws;              // support0, later h
    float* accA  = supA  + NH;      // gcn0 accumulator -> f0
    float* supB  = accA  + NH;      // support1, later l
    float* accB  = supB  + NH;      // gcn1 accumulator -> f1
    float* gacc  = accB  + NH;      // gat accumulator -> feat_feat
    float* el    = gacc  + NH;
    float* er    = el    + N_NODES;
    float* emax  = er    + N_NODES;
    float* denom = emax  + N_NODES;
    float* ebuf  = denom + N_NODES; // per-edge e / ee
    float* scal  = ebuf  + N_EDGES; // [0]=wsum0 [1]=wsum1 [2]=beta0 [3]=beta1

    const int TB = 256;
    int nhBlocks   = (int)((NH + TB - 1) / TB);
    int nodeBlocks = (N_NODES + TB - 1) / TB;
    int edgeBlocks = (N_EDGES + TB - 1) / TB;
    long long scat = (long long)N_EDGES * HID;
    int scatBlocks = (int)((scat + TB - 1) / TB);
    int gemmBlocks = N_NODES / 16;   // 100000 = 16 * 6250, exact

    // 0) init accumulators
    fill_kernel<<<nhBlocks, TB, 0, stream>>>(accA, (long long)NH, 0.f);
    fill_kernel<<<nhBlocks, TB, 0, stream>>>(accB, (long long)NH, 0.f);
    fill_kernel<<<nhBlocks, TB, 0, stream>>>(gacc, (long long)NH, 0.f);
    fill_kernel<<<nodeBlocks, TB, 0, stream>>>(denom, N_NODES, 0.f);
    fill_kernel<<<nodeBlocks, TB, 0, stream>>>(emax, N_NODES, -INFINITY);
    fill_kernel<<<1, 4, 0, stream>>>(scal, 4, 0.f);

    // 1) GCN dense parts (fp32 WMMA)
    gemm_f32_wmma<<<gemmBlocks, 128, 0, stream>>>(attr0, gcn0_W, supA, T0_DIM, nullptr, 0);
    gemm_f32_wmma<<<gemmBlocks, 128, 0, stream>>>(attr1, gcn1_W, supB, T1_DIM, nullptr, 0);

    // 2) GCN sparse aggregation + activation
    gcn_scatter<<<scatBlocks, TB, 0, stream>>>(supA, adj0_src, adj0_dst, adj0_val, accA);
    gcn_scatter<<<scatBlocks, TB, 0, stream>>>(supB, adj1_src, adj1_dst, adj1_val, accB);
    bias_tanh_inplace<<<nhBlocks, TB, 0, stream>>>(accA, gcn0_b);   // f0
    bias_tanh_inplace<<<nhBlocks, TB, 0, stream>>>(accB, gcn1_b);   // f1

    // 3) l = tanh(concat(f0,f1) @ lin1_W + lin1_b)  (writes supB; support1 dead)
    gemm_concat_wmma<<<gemmBlocks, 128, 0, stream>>>(accA, accB, lin1_W, supB, lin1_b);

    // 4) GAT: h = feature @ gat_W (writes supA; support0 dead)
    gemm_f32_wmma<<<gemmBlocks, 128, 0, stream>>>(feature, gat_W, supA, HID, nullptr, 0);
    gat_eler<<<nodeBlocks, TB, 0, stream>>>(supA, gat_al, gat_ar, el, er);
    gat_edge1<<<edgeBlocks, TB, 0, stream>>>(el, er, sim_src, sim_dst, ebuf, emax);
    gat_edge2<<<edgeBlocks, TB, 0, stream>>>(ebuf, sim_dst, emax, denom);
    gat_scatter<<<scatBlocks, TB, 0, stream>>>(supA, sim_src, sim_dst, ebuf, denom, gacc);
    bias_tanh_inplace<<<nhBlocks, TB, 0, stream>>>(gacc, gat_bias); // feat_feat

    // 5) semantic attention: mean-reduced scores -> beta -> blend
    sem_kernel<<<nodeBlocks, TB, 0, stream>>>(supB, gacc, sem_W1, sem_b1, sem_W2, scal);
    beta_kernel<<<1, 1, 0, stream>>>(scal, scal + 2);
    final_kernel<<<nhBlocks, TB, 0, stream>>>(supB, gacc, scal + 2, out);
}